// GCNEncoder_29291676959174
// MI455X (gfx1250) — compile-verified
//
#include <hip/hip_runtime.h>

typedef __attribute__((ext_vector_type(2))) float v2f;
typedef __attribute__((ext_vector_type(8))) float v8f;

// ---------------------------------------------------------------------------
// Utility fills
// ---------------------------------------------------------------------------
__global__ void fill_kernel(float* __restrict__ p, float v, size_t n) {
    size_t i = (size_t)blockIdx.x * blockDim.x + threadIdx.x;
    if (i < n) p[i] = v;
}

__global__ void rsqrt_inplace_kernel(float* __restrict__ p, int n) {
    int i = blockIdx.x * blockDim.x + threadIdx.x;
    if (i < n) p[i] = rsqrtf(p[i]);
}

// deg[dst] += 1 per edge (deg pre-initialized to 1.0 for the self loop)
__global__ void degree_kernel(const long long* __restrict__ dst,
                              float* __restrict__ deg, int nedges) {
    int e = blockIdx.x * blockDim.x + threadIdx.x;
    if (e < nedges) atomicAdd(&deg[dst[e]], 1.0f);
}

// ---------------------------------------------------------------------------
// WMMA fp32 GEMM:  H[nrows x OUT] = X[nrows x 128] * W[128 x OUT]
// Per block: 8 waves, each owns a 16-row strip. W staged TRANSPOSED in LDS
// with a 132-float row stride (bank-conflict-free b64 fragment loads).
// A-fragment layout (16x4 f32): lane<16 -> K=4k,4k+1 ; lane>=16 -> K=4k+2,4k+3
// B-fragment layout (4x16 f32): mirrored over N. C/D: VGPR j -> row j + 8*half.
// ---------------------------------------------------------------------------
template <int OUT>
__global__ __launch_bounds__(256) void gemm_wmma_kernel(
    const float* __restrict__ X, const float* __restrict__ W,
    float* __restrict__ H, int nrows) {
    constexpr int LDSW = 132;  // padded k-stride (floats)
    __shared__ float sWT[OUT * LDSW];  // sWT[n*LDSW + k] = W[k*OUT + n]
    const int t = threadIdx.x;
    for (int i = t; i < OUT * 128; i += 256) {
        int n = i >> 7;    // output column
        int k = i & 127;   // k index
        sWT[n * LDSW + k] = W[k * OUT + n];
    }
    __syncthreads();

    const int wave = t >> 5;
    const int lane = t & 31;
    const int half = lane >> 4;  // 0 or 1
    const int idx  = lane & 15;

    const int strip = blockIdx.x * 8 + wave;  // 16-row strip index
    const int row0  = strip * 16;
    if (row0 >= nrows) return;                // wave-uniform exit, EXEC stays full
    const int row = row0 + idx;

    // Cache this wave's A strip fragments in registers (64 VGPRs).
    v2f afrag[32];
    const float* xr = X + (size_t)row * 128 + 2 * half;
#pragma unroll
    for (int kk = 0; kk < 32; ++kk)
        afrag[kk] = *(const v2f*)(xr + 4 * kk);

#pragma unroll
    for (int ct = 0; ct < OUT / 16; ++ct) {
        v8f acc = {0.f, 0.f, 0.f, 0.f, 0.f, 0.f, 0.f, 0.f};
        const float* wb = sWT + (size_t)(ct * 16 + idx) * LDSW + 2 * half;
#pragma unroll
        for (int kk = 0; kk < 32; ++kk) {
            v2f b = *(const v2f*)(wb + 4 * kk);  // single ds_load_b64
            acc = __builtin_amdgcn_wmma_f32_16x16x4_f32(
                false, afrag[kk], false, b, (short)0, acc, false, false);
        }
        float* o = H + (size_t)(row0 + 8 * half) * OUT + ct * 16 + idx;
#pragma unroll
        for (int j = 0; j < 8; ++j) o[(size_t)j * OUT] = acc[j];
    }
}

// ---------------------------------------------------------------------------
// Edge scatter: one wave per edge, lane-strided features, coalesced
// global_atomic_add_f32 into agg[dst].
// ---------------------------------------------------------------------------
template <int F>
__global__ __launch_bounds__(256) void scatter_kernel(
    const float* __restrict__ H, const long long* __restrict__ src,
    const long long* __restrict__ dst, const float* __restrict__ dis,
    float* __restrict__ agg, int nedges) {
    int gid  = blockIdx.x * blockDim.x + threadIdx.x;
    int edge = gid >> 5;
    int lane = threadIdx.x & 31;
    if (edge >= nedges) return;
    long long s = src[edge];
    long long d = dst[edge];
    float coef = dis[s] * dis[d];
    const float* hs = H + (size_t)s * F;
    float* ad = agg + (size_t)d * F;
#pragma unroll
    for (int i = 0; i < F / 32; ++i) {
        float v = hs[lane + i * 32] * coef;
        atomicAdd(&ad[lane + i * 32], v);
    }
}

// ---------------------------------------------------------------------------
// Layer-1 epilogue: agg = relu(agg + h*dis^2 + b)   (in place)
// ---------------------------------------------------------------------------
template <int F>
__global__ void selfloop_bias_relu_kernel(float* __restrict__ agg,
                                          const float* __restrict__ H,
                                          const float* __restrict__ dis,
                                          const float* __restrict__ b, int n) {
    size_t i = (size_t)blockIdx.x * blockDim.x + threadIdx.x;
    if (i >= (size_t)n * F) return;
    int node = (int)(i / F);
    int f = (int)(i % F);
    float d = dis[node];
    float v = agg[i] + H[i] * d * d + b[f];
    agg[i] = fmaxf(v, 0.f);
}

// ---------------------------------------------------------------------------
// Layer-2 epilogue + row L2 normalize: one wave per node (64 feats = 2/lane)
// ---------------------------------------------------------------------------
__global__ __launch_bounds__(256) void final_norm_kernel(
    const float* __restrict__ agg, const float* __restrict__ H,
    const float* __restrict__ dis, const float* __restrict__ b,
    float* __restrict__ out, int n) {
    int gid  = blockIdx.x * blockDim.x + threadIdx.x;
    int node = gid >> 5;
    int lane = threadIdx.x & 31;
    if (node >= n) return;
    float d  = dis[node];
    float d2 = d * d;
    size_t base = (size_t)node * 64;
    float v0 = agg[base + lane]      + H[base + lane] * d2      + b[lane];
    float v1 = agg[base + 32 + lane] + H[base + 32 + lane] * d2 + b[32 + lane];
    float ss = v0 * v0 + v1 * v1;
#pragma unroll
    for (int off = 16; off > 0; off >>= 1) ss += __shfl_xor(ss, off, 32);
    float norm = fmaxf(sqrtf(ss), 1e-12f);
    float inv  = 1.0f / norm;
    out[base + lane]      = v0 * inv;
    out[base + 32 + lane] = v1 * inv;
}

// ---------------------------------------------------------------------------
// Host launch
// ---------------------------------------------------------------------------
extern "C" void kernel_launch(void* const* d_in, const int* in_sizes, int n_in,
                              void* d_out, int out_size, void* d_ws, size_t ws_size,
                              hipStream_t stream) {
    (void)n_in; (void)out_size; (void)ws_size;
    const float*      x   = (const float*)d_in[0];       // (N,128)
    const long long*  ei  = (const long long*)d_in[1];   // (2,E) int64
    const float*      W1  = (const float*)d_in[2];       // (128,128)
    const float*      b1  = (const float*)d_in[3];       // (128)
    const float*      W2  = (const float*)d_in[4];       // (128,64)
    const float*      b2  = (const float*)d_in[5];       // (64)
    float*            out = (float*)d_out;               // (N,64)

    const int N = in_sizes[0] / 128;
    const int E = in_sizes[1] / 2;
    const long long* src = ei;
    const long long* dst = ei + E;

    // Workspace layout (floats): [dis: N (padded)] [bufA: N*128] [bufB: N*128]
    float* ws   = (float*)d_ws;
    size_t disPad = ((size_t)N + 255) & ~(size_t)255;
    float* dis  = ws;
    float* bufA = ws + disPad;                 // h1, later h2 + agg2
    float* bufB = bufA + (size_t)N * 128;      // agg1, later X2 (in place)

    float* h1   = bufA;
    float* agg1 = bufB;
    float* X2   = bufB;                        // relu'd layer-1 output (in place)
    float* h2   = bufA;                        // N*64
    float* agg2 = bufA + (size_t)N * 64;       // N*64

    const int TPB = 256;
    const int strips = (N + 15) / 16;
    const int gemmBlocks = (strips + 7) / 8;

    // deg = 1.0 (self loop), agg1 = 0
    fill_kernel<<<(N + TPB - 1) / TPB, TPB, 0, stream>>>(dis, 1.0f, (size_t)N);
    fill_kernel<<<(int)(((size_t)N * 128 + TPB - 1) / TPB), TPB, 0, stream>>>(
        agg1, 0.0f, (size_t)N * 128);
    // deg += 1 per in-edge, then dis = rsqrt(deg) in place
    degree_kernel<<<(E + TPB - 1) / TPB, TPB, 0, stream>>>(dst, dis, E);
    rsqrt_inplace_kernel<<<(N + TPB - 1) / TPB, TPB, 0, stream>>>(dis, N);

    // Layer 1
    gemm_wmma_kernel<128><<<gemmBlocks, TPB, 0, stream>>>(x, W1, h1, N);
    scatter_kernel<128><<<(E * 32 + TPB - 1) / TPB, TPB, 0, stream>>>(
        h1, src, dst, dis, agg1, E);
    selfloop_bias_relu_kernel<128>
        <<<(int)(((size_t)N * 128 + TPB - 1) / TPB), TPB, 0, stream>>>(
            agg1, h1, dis, b1, N);

    // Layer 2
    gemm_wmma_kernel<64><<<gemmBlocks, TPB, 0, stream>>>(X2, W2, h2, N);
    fill_kernel<<<(int)(((size_t)N * 64 + TPB - 1) / TPB), TPB, 0, stream>>>(
        agg2, 0.0f, (size_t)N * 64);
    scatter_kernel<64><<<(E * 32 + TPB - 1) / TPB, TPB, 0, stream>>>(
        h2, src, dst, dis, agg2, E);
    final_norm_kernel<<<(N * 32 + TPB - 1) / TPB, TPB, 0, stream>>>(
        agg2, h2, dis, b2, out, N);
}